// ReferenceQwen3MoeSparseMoeBlock_12532714570011
// MI455X (gfx1250) — compile-verified
//
#include <hip/hip_runtime.h>
#include <math.h>

// ---- problem constants (Qwen3 MoE block) ----
#define HID   2048       // hidden size H
#define IMM   768        // moe intermediate I
#define NEXP  32         // experts E
#define TOPK  8
#define NTOK  1024       // T = B*S
#define MTILE 32         // gathered token rows per block (2 WMMA row tiles)
#define TILES (NTOK / MTILE)   // 32

typedef float v2f __attribute__((ext_vector_type(2)));
typedef float v8f __attribute__((ext_vector_type(8)));

__device__ __forceinline__ v8f wmma_f32(v2f a, v2f b, v8f c) {
    // D = A(16x4) * B(4x16) + C(16x16), fp32, wave32
    return __builtin_amdgcn_wmma_f32_16x16x4_f32(false, a, false, b, (short)0, c, false, false);
}

// ===================================================================
// Kernel 1: router. One wave32 per token, lane = expert.
// Writes raw logits to logits_out (2nd tuple output) and normalized
// top-8 (weight, index) to workspace.
// ===================================================================
__global__ __launch_bounds__(256)
void moe_router_kernel(const float* __restrict__ x,      // [T, H]
                       const float* __restrict__ gate_w, // [E, H]
                       float* __restrict__ logits_out,   // [T, E]
                       float* __restrict__ topk_w,       // [T, K]
                       int*   __restrict__ topk_i)       // [T, K]
{
    const int t    = blockIdx.x * 8 + (threadIdx.x >> 5);
    const int lane = threadIdx.x & 31;

    const float* xr = x      + (size_t)t    * HID;
    const float* gr = gate_w + (size_t)lane * HID;

    float acc = 0.0f;
    #pragma unroll 4
    for (int h = 0; h < HID; h += 4) {
        float4 xv = *(const float4*)(xr + h);
        float4 gv = *(const float4*)(gr + h);
        acc += xv.x * gv.x + xv.y * gv.y + xv.z * gv.z + xv.w * gv.w;
    }
    logits_out[t * NEXP + lane] = acc;   // raw router logits (pre-softmax)

    // softmax across the 32 lanes (= experts)
    float m = acc;
    #pragma unroll
    for (int off = 16; off >= 1; off >>= 1) m = fmaxf(m, __shfl_xor(m, off));
    float p = expf(acc - m);
    float s = p;
    #pragma unroll
    for (int off = 16; off >= 1; off >>= 1) s += __shfl_xor(s, off);
    float prob = p / s;

    // iterative top-8, tie-break to lowest index (matches jax.lax.top_k)
    float my = prob;
    float wk[TOPK]; int ik[TOPK]; float wsum = 0.0f;
    #pragma unroll
    for (int k = 0; k < TOPK; ++k) {
        float bv = my; int bi = lane;
        #pragma unroll
        for (int off = 16; off >= 1; off >>= 1) {
            float ov = __shfl_xor(bv, off);
            int   oi = __shfl_xor(bi, off);
            if (ov > bv || (ov == bv && oi < bi)) { bv = ov; bi = oi; }
        }
        wk[k] = bv; ik[k] = bi; wsum += bv;
        if (lane == bi) my = -1e30f;
    }
    if (lane == 0) {
        float inv = 1.0f / wsum;     // norm_topk_prob=True
        #pragma unroll
        for (int k = 0; k < TOPK; ++k) {
            topk_w[t * TOPK + k] = wk[k] * inv;
            topk_i[t * TOPK + k] = ik[k];
        }
    }
}

// ===================================================================
// Kernel 2: deterministic ordered token-gather per expert.
// One wave32 per expert; ballot + prefix popcount compaction.
// pairs[] entries are (t*8 + k) so they directly index y_stage slots.
// ===================================================================
__global__ __launch_bounds__(32)
void moe_build_lists_kernel(const int* __restrict__ topk_i,
                            int* __restrict__ counts,   // [E]
                            int* __restrict__ pairs)    // [E, T]
{
    const int e    = blockIdx.x;
    const int lane = threadIdx.x;
    int base = 0;
    for (int t0 = 0; t0 < NTOK; t0 += 32) {
        const int t = t0 + lane;
        int kk = -1;
        #pragma unroll
        for (int k = 0; k < TOPK; ++k)
            if (topk_i[t * TOPK + k] == e) kk = k;
        unsigned mask = (unsigned)__ballot(kk >= 0);
        if (kk >= 0) {
            int pos = __popc(mask & ((1u << lane) - 1u));
            pairs[e * NTOK + base + pos] = t * TOPK + kk;
        }
        base += __popc(mask);
    }
    if (lane == 0) counts[e] = base;
}

// ===================================================================
// Kernel 3: fused expert SwiGLU MLP on gathered rows.
// Block = 256 threads = 8 waves; handles (expert e, 32 gathered rows).
// Two 16-row WMMA tiles per wave per column tile -> each weight (B)
// fragment feeds 2 row tiles (4 loads -> 4 WMMAs per K-step).
// Phase 1: G/U = X*Wg^T / X*Wu^T (v_wmma_f32_16x16x4_f32), SiLU fused,
//          ACT[32 x 768] staged in LDS (96 KB, well under 320 KB/WGP).
// Phase 2: Y = ACT * Wd^T, stored to per-(t,k) staging (no atomics).
// ===================================================================
__global__ __launch_bounds__(256)
void moe_expert_kernel(const float* __restrict__ x,         // [T, H]
                       const float* __restrict__ gate_proj, // [E, I, H]
                       const float* __restrict__ up_proj,   // [E, I, H]
                       const float* __restrict__ down_proj, // [E, H, I]
                       const int*   __restrict__ counts,
                       const int*   __restrict__ pairs,
                       float* __restrict__ y_stage)         // [T*K, H]
{
    __shared__ float ACTs[MTILE * IMM];   // 96 KB
    __shared__ int   pair_lds[MTILE];

    const int e    = blockIdx.x / TILES;
    const int tile = blockIdx.x % TILES;
    const int cnt  = counts[e];
    const int row0 = tile * MTILE;
    if (row0 >= cnt) return;              // uniform early exit (EXEC stays full)

    const int tid = threadIdx.x;
    if (tid < MTILE) {
        int r = min(row0 + tid, cnt - 1); // clamp; garbage rows masked on store
        pair_lds[tid] = pairs[e * NTOK + r];
    }
    __syncthreads();

    const int wave = tid >> 5;
    const int lane = tid & 31;
    const int half = lane >> 4;           // 0: K0/K1 lanes, 1: K2/K3 lanes
    const int l15  = lane & 15;

    // A-fragment source rows for this lane (gathered tokens, two row tiles)
    const int ta0 = pair_lds[l15]      >> 3;
    const int ta1 = pair_lds[16 + l15] >> 3;
    const float* aG0 = x + (size_t)ta0 * HID + half * 2;
    const float* aG1 = x + (size_t)ta1 * HID + half * 2;

    // ---------------- Phase 1: gate/up GEMMs + SiLU ----------------
    for (int ni = wave; ni < IMM / 16; ni += 8) {
        const size_t wrow = ((size_t)e * IMM + ni * 16 + l15) * HID + half * 2;
        const float* bg = gate_proj + wrow;
        const float* bu = up_proj   + wrow;
        v8f cg0 = {}; v8f cu0 = {};
        v8f cg1 = {}; v8f cu1 = {};
        #pragma unroll 4
        for (int kc = 0; kc < HID; kc += 4) {
            v2f a0  = *(const v2f*)(aG0 + kc);
            v2f a1  = *(const v2f*)(aG1 + kc);
            v2f bgv = *(const v2f*)(bg + kc);
            v2f buv = *(const v2f*)(bu + kc);
            cg0 = wmma_f32(a0, bgv, cg0);
            cg1 = wmma_f32(a1, bgv, cg1);
            cu0 = wmma_f32(a0, buv, cu0);
            cu1 = wmma_f32(a1, buv, cu1);
        }
        #pragma unroll
        for (int j = 0; j < 8; ++j) {
            int row = j + half * 8;                        // C-tile M row
            float g0 = cg0[j], u0 = cu0[j];
            ACTs[row * IMM + ni * 16 + l15] =
                (g0 / (1.0f + expf(-g0))) * u0;            // silu(g)*u
            float g1 = cg1[j], u1 = cu1[j];
            ACTs[(16 + row) * IMM + ni * 16 + l15] =
                (g1 / (1.0f + expf(-g1))) * u1;
        }
    }
    __syncthreads();

    // ---------------- Phase 2: down projection ----------------
    const float* aL0 = &ACTs[l15 * IMM + half * 2];
    const float* aL1 = &ACTs[(16 + l15) * IMM + half * 2];
    for (int ni = wave; ni < HID / 16; ni += 8) {
        const float* bd = down_proj + ((size_t)e * HID + ni * 16 + l15) * IMM + half * 2;
        v8f c0 = {}; v8f c1 = {};
        #pragma unroll 4
        for (int kc = 0; kc < IMM; kc += 4) {
            v2f a0 = *(const v2f*)(aL0 + kc);
            v2f a1 = *(const v2f*)(aL1 + kc);
            v2f b  = *(const v2f*)(bd + kc);
            c0 = wmma_f32(a0, b, c0);
            c1 = wmma_f32(a1, b, c1);
        }
        #pragma unroll
        for (int j = 0; j < 8; ++j) {
            int row = j + half * 8;
            if (row0 + row < cnt) {
                int p = pair_lds[row];                      // == t*8 + k
                y_stage[(size_t)p * HID + ni * 16 + l15] = c0[j];
            }
            int row2 = 16 + row;
            if (row0 + row2 < cnt) {
                int p = pair_lds[row2];
                y_stage[(size_t)p * HID + ni * 16 + l15] = c1[j];
            }
        }
    }
}

// ===================================================================
// Kernel 4: weighted combine over the 8 routed experts per token.
// ===================================================================
__global__ __launch_bounds__(256)
void moe_combine_kernel(const float* __restrict__ y_stage, // [T*K, H]
                        const float* __restrict__ topk_w,  // [T, K]
                        float* __restrict__ out)           // [T, H]
{
    const size_t idx = (size_t)blockIdx.x * 256 + threadIdx.x;  // T*H threads
    const int t = (int)(idx >> 11);       // / 2048
    const int h = (int)(idx & 2047);
    float s = 0.0f;
    #pragma unroll
    for (int k = 0; k < TOPK; ++k) {
        int p = t * TOPK + k;
        s += topk_w[p] * y_stage[(size_t)p * HID + h];
    }
    out[idx] = s;
}

// ===================================================================
extern "C" void kernel_launch(void* const* d_in, const int* in_sizes, int n_in,
                              void* d_out, int out_size, void* d_ws, size_t ws_size,
                              hipStream_t stream)
{
    const float* x         = (const float*)d_in[0];   // [1,1024,2048]
    const float* gate_w    = (const float*)d_in[1];   // [32,2048]
    const float* gate_proj = (const float*)d_in[2];   // [32,768,2048]
    const float* up_proj   = (const float*)d_in[3];   // [32,768,2048]
    const float* down_proj = (const float*)d_in[4];   // [32,2048,768]

    float* out_final  = (float*)d_out;                // T*H
    float* out_logits = (float*)d_out + (size_t)NTOK * HID;  // T*E (2nd tuple output)

    // workspace carve (~67.3 MB total)
    char* p = (char*)d_ws;
    float* topk_w = (float*)p;  p += (size_t)NTOK * TOPK * sizeof(float);
    int*   topk_i = (int*)p;    p += (size_t)NTOK * TOPK * sizeof(int);
    int*   counts = (int*)p;    p += 256;
    int*   pairs  = (int*)p;    p += (size_t)NEXP * NTOK * sizeof(int);
    p = (char*)(((uintptr_t)p + 255) & ~(uintptr_t)255);
    float* ystage = (float*)p;  // [T*K, H] = 67,108,864 bytes

    moe_router_kernel<<<NTOK / 8, 256, 0, stream>>>(x, gate_w, out_logits, topk_w, topk_i);
    moe_build_lists_kernel<<<NEXP, 32, 0, stream>>>(topk_i, counts, pairs);
    moe_expert_kernel<<<NEXP * TILES, 256, 0, stream>>>(x, gate_proj, up_proj, down_proj,
                                                        counts, pairs, ystage);
    moe_combine_kernel<<<(NTOK * HID) / 256, 256, 0, stream>>>(ystage, topk_w, out_final);
}